// HandCraftedFeatures_10883447128428
// MI455X (gfx1250) — compile-verified
//
#include <hip/hip_runtime.h>
#include <hip/hip_bf16.h>
#include <stdint.h>
#include <math.h>

#define BATCH 32
#define HW    512
#define CELLP 16          // pixels per cell
#define ORI   6
#define NC    32          // cells per dim (512/16)
#define NB    31          // blocks per dim
#define STRIP 18          // 16 rows + 2 halo rows

// bin = floor((atan2(gr,gc) deg mod 180)/30) via half-plane tests (no atan2).
__device__ __forceinline__ int hog_bin(float gr_, float gc_)
{
    // fold angle into [0,180): negate both if gr<0, or gr==0 && gc<0
    if (gr_ < 0.0f || (gr_ == 0.0f && gc_ < 0.0f)) { gr_ = -gr_; gc_ = -gc_; }
    const float S = 0.5f;                    // sin30 = cos60
    const float C = 0.8660254037844387f;     // cos30 = sin60
    int b = 0;
    b += (S * gc_ - C * gr_ <= 0.0f);        // theta >= 30
    b += (C * gc_ - S * gr_ <= 0.0f);        // theta >= 60
    b += (gc_ <= 0.0f);                      // theta >= 90
    b += (C * gc_ + S * gr_ <= 0.0f);        // theta >= 120
    b += (S * gc_ + C * gr_ <= 0.0f);        // theta >= 150
    return b;                                // 0..5 (ties are measure-zero)
}

// ---------------------------------------------------------------------------
// Kernel 1: per-(image, cell-row) strip -> 32 cells x 6 orientation bins.
// One async burst stages all 3 channel strips (18x512 f32 each) into LDS via
// GLOBAL_LOAD_ASYNC_TO_LDS_B128 (ASYNCcnt), luma is fused in-place over the
// ch-0 region (float4 LDS ops), gradients roll through registers, histograms
// accumulate with ds_add_f32 atomics.
// ---------------------------------------------------------------------------
__global__ __launch_bounds__(256)
void hog_cells_kernel(const float* __restrict__ x, float* __restrict__ cells)
{
    __shared__ float s_raw[3 * STRIP * HW];   // 110,592 B: ch0 region -> gray
    __shared__ float s_hist[NC * ORI];

    const int tid = threadIdx.x;
    const int cr  = blockIdx.x & 31;          // cell row
    const int img = blockIdx.x >> 5;          // image
    const int r0  = cr * CELLP - 1;           // global row of strip row 0

    if (tid < NC * ORI) s_hist[tid] = 0.0f;

    // ---- Phase A: async-stage all 3 channel strips (valid rows only) ----
    const uint32_t lds_base = (uint32_t)(uintptr_t)(&s_raw[0]);
    const int kmin = (cr == 0)      ? 1 : 0;
    const int kmax = (cr == NC - 1) ? (STRIP - 2) : (STRIP - 1);
    const int nchunk = (kmax - kmin + 1) * (HW / 4);   // 16B chunks per channel

    #pragma unroll
    for (int ch = 0; ch < 3; ++ch) {
        const uint64_t gbase =
            (uint64_t)(uintptr_t)x + (uint64_t)((size_t)(img * 3 + ch) * HW * HW) * 4u;
        for (int i = tid; i < nchunk; i += 256) {
            const int k      = kmin + (i >> 7);              // strip row
            const uint32_t w = (uint32_t)(i & 127) * 16u;    // byte within row
            const uint32_t ldsoff = lds_base + (uint32_t)(ch * STRIP + k) * (HW * 4u) + w;
            const uint32_t goff   = (uint32_t)(r0 + k) * (HW * 4u) + w;
            asm volatile("global_load_async_to_lds_b128 %0, %1, %2"
                         :: "v"(ldsoff), "v"(goff), "s"(gbase)
                         : "memory");
        }
    }
    asm volatile("s_wait_asynccnt 0" ::: "memory");
    __syncthreads();

    // ---- Phase B: fuse luma in place over the ch-0 region (float4) ----
    {
        float4* raw4 = (float4*)s_raw;
        const int n4 = (STRIP * HW) / 4;      // 2304
        for (int i = tid; i < n4; i += 256) {
            const float4 r = raw4[i];
            const float4 g = raw4[n4 + i];
            const float4 b = raw4[2 * n4 + i];
            float4 o;
            o.x = fmaf(0.114f, b.x, fmaf(0.587f, g.x, 0.299f * r.x));
            o.y = fmaf(0.114f, b.y, fmaf(0.587f, g.y, 0.299f * r.y));
            o.z = fmaf(0.114f, b.z, fmaf(0.587f, g.z, 0.299f * r.z));
            o.w = fmaf(0.114f, b.w, fmaf(0.587f, g.w, 0.299f * r.w));
            raw4[i] = o;
        }
    }
    __syncthreads();

    // ---- Phase C: gradients + binned histogram (rolling row registers) ----
    // Thread owns columns {c0, c0+1}; gray plane lives in s_raw[0 .. 18*512).
    const float* gray = s_raw;
    const int c0   = tid * 2;
    const int cm1  = (c0 == 0)        ? 0       : c0 - 1;    // clamped halo
    const int cp2  = (c0 == HW - 2)   ? HW - 1  : c0 + 2;    // clamped halo
    const bool colv0 = (c0 >= 1);            // gcol valid for column c0
    const bool colv1 = (c0 + 1 <= HW - 2);   // gcol valid for column c0+1

    float acc[ORI] = {0.f, 0.f, 0.f, 0.f, 0.f, 0.f};

    // up2: centers of row k-1; cur: {c0-1, c0, c0+1, c0+2} of row k
    float2 up = *(const float2*)&gray[0 * HW + c0];          // strip row 0
    float2 cc2 = *(const float2*)&gray[1 * HW + c0];
    float cur_m1 = gray[1 * HW + cm1];
    float cur_p0 = cc2.x, cur_p1 = cc2.y;
    float cur_p2 = gray[1 * HW + cp2];

    for (int rr = 0; rr < CELLP; ++rr) {
        const int gr = cr * CELLP + rr;       // global row; strip row k = rr+1
        const int kd = rr + 2;                // strip row k+1
        const float2 dn = *(const float2*)&gray[kd * HW + c0];
        const bool rowv = (gr >= 1) && (gr <= HW - 2);

        const float grow0 = rowv  ? (dn.x - up.x) : 0.0f;
        const float grow1 = rowv  ? (dn.y - up.y) : 0.0f;
        const float gcol0 = colv0 ? (cur_p1 - cur_m1) : 0.0f;
        const float gcol1 = colv1 ? (cur_p2 - cur_p0) : 0.0f;

        const float mag0 = sqrtf(grow0 * grow0 + gcol0 * gcol0);
        const float mag1 = sqrtf(grow1 * grow1 + gcol1 * gcol1);
        const int b0 = hog_bin(grow0, gcol0);
        const int b1 = hog_bin(grow1, gcol1);
        #pragma unroll
        for (int o = 0; o < ORI; ++o) {
            acc[o] += (b0 == o) ? mag0 : 0.0f;
            acc[o] += (b1 == o) ? mag1 : 0.0f;
        }

        // roll rows: k-1 <- k centers; k <- k+1 (extend halos)
        up.x = cur_p0; up.y = cur_p1;
        cur_m1 = gray[kd * HW + cm1];
        cur_p0 = dn.x; cur_p1 = dn.y;
        cur_p2 = gray[kd * HW + cp2];
    }

    const int cellc = tid >> 3;               // cell column (2t / 16)
    #pragma unroll
    for (int o = 0; o < ORI; ++o)
        atomicAdd(&s_hist[cellc * ORI + o], acc[o]);   // ds_add_f32
    __syncthreads();

    if (tid < NC * ORI)
        cells[(size_t)(img * NC + cr) * (NC * ORI) + tid] =
            s_hist[tid] * (1.0f / (CELLP * CELLP));
}

// ---------------------------------------------------------------------------
// Kernel 2: L2-Hys normalization of 2x2-cell blocks, one thread per block.
// Output layout matches reference flatten: (B, 31, 31, 2, 2, 6).
// ---------------------------------------------------------------------------
__global__ __launch_bounds__(256)
void hog_blocks_kernel(const float* __restrict__ cells, float* __restrict__ out)
{
    const int idx   = blockIdx.x * blockDim.x + threadIdx.x;
    const int total = BATCH * NB * NB;
    if (idx >= total) return;

    const int img = idx / (NB * NB);
    const int rem = idx - img * (NB * NB);
    const int br  = rem / NB;
    const int bc  = rem - br * NB;

    float v[2 * 2 * ORI];
    float ss = 0.0f;
    #pragma unroll
    for (int i = 0; i < 2; ++i)
        #pragma unroll
        for (int j = 0; j < 2; ++j)
            #pragma unroll
            for (int o = 0; o < ORI; ++o) {
                const float t =
                    cells[(((size_t)img * NC + (br + i)) * NC + (bc + j)) * ORI + o];
                v[(i * 2 + j) * ORI + o] = t;
                ss += t * t;
            }

    const float eps2 = 1e-10f;               // EPS*EPS, EPS = 1e-5
    const float inv1 = 1.0f / sqrtf(ss + eps2);

    float ss2 = 0.0f;
    #pragma unroll
    for (int q = 0; q < 2 * 2 * ORI; ++q) {
        const float t = fminf(v[q] * inv1, 0.2f);
        v[q] = t;
        ss2 += t * t;
    }
    const float inv2 = 1.0f / sqrtf(ss2 + eps2);

    const size_t ob = (size_t)idx * (2 * 2 * ORI);
    #pragma unroll
    for (int q = 0; q < 2 * 2 * ORI; ++q)
        out[ob + q] = v[q] * inv2;
}

// ---------------------------------------------------------------------------
extern "C" void kernel_launch(void* const* d_in, const int* in_sizes, int n_in,
                              void* d_out, int out_size, void* d_ws, size_t ws_size,
                              hipStream_t stream)
{
    const float* x = (const float*)d_in[0];
    float* cells   = (float*)d_ws;            // 32*32*32*6 f32 = 786,432 B
    float* out     = (float*)d_out;           // 32*31*31*24 f32

    hog_cells_kernel<<<dim3(BATCH * NC), dim3(256), 0, stream>>>(x, cells);

    const int total_blocks = BATCH * NB * NB; // 30,752
    hog_blocks_kernel<<<dim3((total_blocks + 255) / 256), dim3(256), 0, stream>>>(cells, out);
}